// InterSetLayer_30640296689893
// MI455X (gfx1250) — compile-verified
//
#include <hip/hip_runtime.h>

// InterSetLayer for MI455X (gfx1250, wave32, WMMA + async global->LDS).
// f32-exact GEMMs via V_WMMA_F32_16X16X4_F32 with B = [W_q | W_v] packed into
// the 16 N-columns; positional branch folded algebraically into the epilogue.
// x-tile staged with GLOBAL_LOAD_ASYNC_TO_LDS_B128 (ASYNCcnt), overlapping the
// per-row BN/relu work with the copy.

#define OUT_P 64
#define MID_P 8
#define NS_P  32
#define EPSV  1e-5f
#define LDSW  68   // padded LDS row (floats), 16B-aligned rows, avoids bank conflicts

typedef __attribute__((ext_vector_type(2))) float v2f;
typedef __attribute__((ext_vector_type(8))) float v8f;

__device__ __forceinline__ unsigned int f2ord(float f) {
    unsigned int u = __float_as_uint(f);
    return (u & 0x80000000u) ? ~u : (u | 0x80000000u);
}
__device__ __forceinline__ float ord2f(unsigned int o) {
    unsigned int u = (o & 0x80000000u) ? (o & 0x7FFFFFFFu) : ~o;
    return __uint_as_float(u);
}

// ---------------- init segment accumulators ----------------
__global__ __launch_bounds__(256) void init_kernel(unsigned int* __restrict__ gmax,
                                                   float* __restrict__ gsum,
                                                   float* __restrict__ gvsum, int n) {
    int i = blockIdx.x * 256 + threadIdx.x;
    if (i < n) {
        gmax[i]  = 0x007FFFFFu;   // ord(-inf)
        gsum[i]  = 0.f;
        gvsum[i] = 0.f;
    }
}

// ---------------- BN batch-stats, pass 1 (per-block partials) ----------------
__global__ __launch_bounds__(256) void bn_partial_kernel(const float* __restrict__ translation,
                                                         const float* __restrict__ W_p1,
                                                         float* __restrict__ partials, int N) {
    __shared__ float red[256 * 4];
    const float w00 = W_p1[0], w01 = W_p1[1], w10 = W_p1[2], w11 = W_p1[3];
    float s0 = 0.f, s1 = 0.f, s2 = 0.f, s3 = 0.f;
    int base = blockIdx.x * 2048 + threadIdx.x;
    #pragma unroll
    for (int j = 0; j < 8; ++j) {
        int i = base + j * 256;
        if (i < N) {
            float tr0 = translation[2 * i + 0], tr1 = translation[2 * i + 1];
            float t0 = w00 * tr0 + w01 * tr1;
            float t1 = w10 * tr0 + w11 * tr1;
            s0 += t0; s1 += t1; s2 += t0 * t0; s3 += t1 * t1;
        }
    }
    int t = threadIdx.x;
    red[t * 4 + 0] = s0; red[t * 4 + 1] = s1; red[t * 4 + 2] = s2; red[t * 4 + 3] = s3;
    __syncthreads();
    for (int off = 128; off > 0; off >>= 1) {
        if (t < off) {
            #pragma unroll
            for (int c = 0; c < 4; ++c) red[t * 4 + c] += red[(t + off) * 4 + c];
        }
        __syncthreads();
    }
    if (t < 4) partials[blockIdx.x * 4 + t] = red[t];
}

// ---------------- BN finalize + tiny folded-weight precompute ----------------
// params[0..1]=scale, [2..3]=shift, [4..19]=Wqp(8x2)=W_q@W_p2, [20..27]=b_q+W_q@b_p2
__global__ __launch_bounds__(64) void bn_finalize_kernel(const float* __restrict__ partials, int nparts,
                                                         const float* __restrict__ gamma,
                                                         const float* __restrict__ beta,
                                                         const float* __restrict__ W_q,
                                                         const float* __restrict__ W_p2,
                                                         const float* __restrict__ b_q,
                                                         const float* __restrict__ b_p2,
                                                         float* __restrict__ params, int N) {
    __shared__ float red[64 * 4];
    int t = threadIdx.x;
    float s0 = 0.f, s1 = 0.f, s2 = 0.f, s3 = 0.f;
    for (int p = t; p < nparts; p += 64) {
        s0 += partials[p * 4 + 0]; s1 += partials[p * 4 + 1];
        s2 += partials[p * 4 + 2]; s3 += partials[p * 4 + 3];
    }
    red[t * 4 + 0] = s0; red[t * 4 + 1] = s1; red[t * 4 + 2] = s2; red[t * 4 + 3] = s3;
    __syncthreads();
    for (int off = 32; off > 0; off >>= 1) {
        if (t < off) {
            #pragma unroll
            for (int c = 0; c < 4; ++c) red[t * 4 + c] += red[(t + off) * 4 + c];
        }
        __syncthreads();
    }
    if (t < 2) {
        float invN  = 1.0f / (float)N;
        float mean  = red[t] * invN;
        float var   = red[2 + t] * invN - mean * mean;   // biased var (torch BN)
        float scale = gamma[t] * rsqrtf(var + EPSV);
        params[t]     = scale;
        params[2 + t] = beta[t] - mean * scale;
    }
    if (t < 16) {                       // Wqp[c][k] = sum_j W_q[c][j] * W_p2[j][k]
        int c = t >> 1, k = t & 1;
        float acc = 0.f;
        for (int j = 0; j < OUT_P; ++j) acc += W_q[c * OUT_P + j] * W_p2[j * 2 + k];
        params[4 + t] = acc;
    } else if (t < 24) {                // bq2[c] = b_q[c] + sum_j W_q[c][j]*b_p2[j]
        int c = t - 16;
        float acc = b_q[c];
        for (int j = 0; j < OUT_P; ++j) acc += W_q[c * OUT_P + j] * b_p2[j];
        params[20 + c] = acc;
    }
}

// ---------------- fused features + q/v GEMM (WMMA f32) + segment-max ----------------
// One block (2 waves) per set: wave w handles rows [16w, 16w+16).
// D(16x16) = Xtile(16x64) x B(64x16), B columns 0..7 = W_q rows, 8..15 = W_v rows.
__global__ __launch_bounds__(64) void qv_wmma_kernel(const float* __restrict__ outputs,
                                                     const float* __restrict__ translation,
                                                     const int*   __restrict__ indexes,
                                                     const float* __restrict__ W_q,
                                                     const float* __restrict__ W_v,
                                                     const float* __restrict__ W_p1,
                                                     const float* __restrict__ b_v,
                                                     const float* __restrict__ params,
                                                     float* __restrict__ feat_out,
                                                     unsigned int* __restrict__ gmax_ord,
                                                     float* __restrict__ qbuf,
                                                     float* __restrict__ vbuf) {
    __shared__ float xs[NS_P * LDSW];
    __shared__ float rt[NS_P * 2];
    __shared__ int   sidx[NS_P];

    const int s    = blockIdx.x;
    const int tid  = threadIdx.x;
    const int lane = tid & 31;
    const int wave = tid >> 5;

    // Stage the 32x64 x-tile into LDS via the CDNA5 async global->LDS path
    // (no VGPR round-trip; tracked by ASYNCcnt). Padded rows of LDSW floats.
    const float* xbase = outputs + (size_t)s * (NS_P * OUT_P);
    #pragma unroll
    for (int j = 0; j < 8; ++j) {
        int l   = tid + 64 * j;        // float4 index within tile, 0..511
        int row = l >> 4;
        int cv  = l & 15;
        unsigned int lds_addr = (unsigned int)(size_t)&xs[row * LDSW + cv * 4];
        unsigned int goff     = (unsigned int)(l * 16);   // byte offset from xbase
        asm volatile("global_load_async_to_lds_b128 %0, %1, %2"
                     :
                     : "v"(lds_addr), "v"(goff), "s"(xbase)
                     : "memory");
    }

    // Overlap with the copy: per-row relu(bn(t @ W_p1^T)) and segment index.
    if (tid < NS_P) {
        int ri = s * NS_P + tid;
        float tr0 = translation[2 * ri + 0];
        float tr1 = translation[2 * ri + 1];
        float t0 = W_p1[0] * tr0 + W_p1[1] * tr1;
        float t1 = W_p1[2] * tr0 + W_p1[3] * tr1;
        t0 = t0 * params[0] + params[2];
        t1 = t1 * params[1] + params[3];
        rt[2 * tid + 0] = fmaxf(t0, 0.f);
        rt[2 * tid + 1] = fmaxf(t1, 0.f);
        sidx[tid] = indexes[ri];
    }

    // B fragments: lane owns column col; K layout: lanes 0-15 -> K%4 in {0,1},
    // lanes 16-31 -> K%4 in {2,3} (per ISA 32-bit A/B 16x4 striping).
    const int col  = lane & 15;
    const int koff = (lane < 16) ? 0 : 2;
    const float* wrow = (col < MID_P) ? (W_q + col * OUT_P) : (W_v + (col - MID_P) * OUT_P);
    v2f bfrag[16];
    #pragma unroll
    for (int kk = 0; kk < 16; ++kk) {
        bfrag[kk].x = wrow[4 * kk + koff + 0];
        bfrag[kk].y = wrow[4 * kk + koff + 1];
    }

    // Wait for this wave's async LDS writes, then barrier for the whole tile.
    asm volatile("s_wait_asynccnt 0x0" ::: "memory");
    __syncthreads();

    // features[s][c] = sum over 32 samples (thread c owns one channel).
    {
        float acc = 0.f;
        #pragma unroll 8
        for (int n = 0; n < NS_P; ++n) acc += xs[n * LDSW + tid];
        feat_out[(size_t)s * OUT_P + tid] = acc;
    }

    // 16 K-steps of V_WMMA_F32_16X16X4_F32 accumulate the full K=64 GEMM in f32.
    const int m0   = wave * 16;
    const int arow = m0 + (lane & 15);
    v8f d = {0.f, 0.f, 0.f, 0.f, 0.f, 0.f, 0.f, 0.f};
    #pragma unroll
    for (int kk = 0; kk < 16; ++kk) {
        v2f a;
        a.x = xs[arow * LDSW + 4 * kk + koff + 0];
        a.y = xs[arow * LDSW + 4 * kk + koff + 1];
        d = __builtin_amdgcn_wmma_f32_16x16x4_f32(false, a, false, bfrag[kk],
                                                  (short)0, d, false, false);
    }

    // Epilogue: cols 0..7 -> q (+ folded positional branch, + segment max),
    //           cols 8..15 -> v (+ bias). D row = r + 8*(lane/16).
    float wqp0 = 0.f, wqp1 = 0.f, bq2 = 0.f, bvc = 0.f;
    if (col < MID_P) {
        wqp0 = params[4 + 2 * col + 0];
        wqp1 = params[4 + 2 * col + 1];
        bq2  = params[20 + col];
    } else {
        bvc = b_v[col - MID_P];
    }
    const int rbase = m0 + 8 * (lane >> 4);
    #pragma unroll
    for (int r = 0; r < 8; ++r) {
        int grow = rbase + r;                    // row within set
        size_t i = (size_t)s * NS_P + grow;      // global row
        if (col < MID_P) {
            float qv = d[r] + rt[2 * grow] * wqp0 + rt[2 * grow + 1] * wqp1 + bq2;
            qbuf[i * MID_P + col] = qv;
            atomicMax(&gmax_ord[(size_t)sidx[grow] * MID_P + col], f2ord(qv));
        } else {
            vbuf[i * MID_P + (col - MID_P)] = d[r] + bvc;
        }
    }
}

// ---------------- exp + segment accumulate ----------------
__global__ __launch_bounds__(256) void softmax_accum_kernel(const int* __restrict__ indexes,
                                                            const float* __restrict__ qbuf,
                                                            const float* __restrict__ vbuf,
                                                            const unsigned int* __restrict__ gmax,
                                                            float* __restrict__ gsum,
                                                            float* __restrict__ gvsum, int total) {
    int g = blockIdx.x * 256 + threadIdx.x;      // over N*8
    if (g >= total) return;
    int i = g >> 3, c = g & 7;
    int seg = indexes[i];
    float e = expf(qbuf[g] - ord2f(gmax[(size_t)seg * MID_P + c]));
    atomicAdd(&gsum[(size_t)seg * MID_P + c], e);
    atomicAdd(&gvsum[(size_t)seg * MID_P + c], e * vbuf[g]);
}

// ---------------- out = features + tile(gvsum/gsum, 8) ----------------
__global__ __launch_bounds__(256) void out_finalize_kernel(const float* __restrict__ gsum,
                                                           const float* __restrict__ gvsum,
                                                           float* __restrict__ out, int total) {
    int g = blockIdx.x * 256 + threadIdx.x;      // over SIZE*64
    if (g < total) {
        int s = g >> 6, m = g & 7;               // (c % 64) % 8 == g & 7
        out[g] += gvsum[s * MID_P + m] / gsum[s * MID_P + m];
    }
}

extern "C" void kernel_launch(void* const* d_in, const int* in_sizes, int n_in,
                              void* d_out, int out_size, void* d_ws, size_t ws_size,
                              hipStream_t stream) {
    const float* outputs     = (const float*)d_in[0];
    const float* translation = (const float*)d_in[1];
    const int*   indexes     = (const int*)d_in[2];
    const float* W_q         = (const float*)d_in[3];
    const float* b_q         = (const float*)d_in[4];
    const float* W_v         = (const float*)d_in[5];
    const float* b_v         = (const float*)d_in[6];
    const float* W_p1        = (const float*)d_in[7];
    const float* bn_gamma    = (const float*)d_in[8];
    const float* bn_beta     = (const float*)d_in[9];
    const float* W_p2        = (const float*)d_in[10];
    const float* b_p2        = (const float*)d_in[11];
    float* out = (float*)d_out;

    const int N     = in_sizes[2];        // SIZE * NS
    const int SIZE_ = out_size / OUT_P;   // number of sets
    (void)n_in; (void)ws_size;

    // Workspace layout (floats).
    float* ws       = (float*)d_ws;
    float* params   = ws;                                   // 32
    float* partials = ws + 32;                              // <= 8160
    unsigned int* gmax = (unsigned int*)(ws + 8192);        // SIZE*8
    float* gsum  = ws + 8192 + (size_t)SIZE_ * MID_P;       // SIZE*8
    float* gvsum = gsum + (size_t)SIZE_ * MID_P;            // SIZE*8
    float* qbuf  = gvsum + (size_t)SIZE_ * MID_P;           // N*8
    float* vbuf  = qbuf + (size_t)N * MID_P;                // N*8

    const int nseg8 = SIZE_ * MID_P;
    init_kernel<<<(nseg8 + 255) / 256, 256, 0, stream>>>(gmax, gsum, gvsum, nseg8);

    const int npart = (N + 2047) / 2048;
    bn_partial_kernel<<<npart, 256, 0, stream>>>(translation, W_p1, partials, N);
    bn_finalize_kernel<<<1, 64, 0, stream>>>(partials, npart, bn_gamma, bn_beta,
                                             W_q, W_p2, b_q, b_p2, params, N);

    qv_wmma_kernel<<<SIZE_, 64, 0, stream>>>(outputs, translation, indexes, W_q, W_v,
                                             W_p1, b_v, params, out, gmax, qbuf, vbuf);

    const int nq = N * MID_P;
    softmax_accum_kernel<<<(nq + 255) / 256, 256, 0, stream>>>(indexes, qbuf, vbuf,
                                                               gmax, gsum, gvsum, nq);

    out_finalize_kernel<<<(out_size + 255) / 256, 256, 0, stream>>>(gsum, gvsum, out,
                                                                    out_size);
}